// NMS_19430432047285
// MI455X (gfx1250) — compile-verified
//
#include <hip/hip_runtime.h>
#include <stdint.h>

typedef __attribute__((ext_vector_type(2))) float v2f;
typedef __attribute__((ext_vector_type(8))) float v8f;

#define TRESH 0.5f
#define NPAD 8192            // power-of-two pad for bitonic sort (N=8000 <= 8192)
#define SORT_THREADS 1024
#define NW 256               // padded u32 words per mask row (256*32 = 8192 bits)

// ---------------------------------------------------------------------------
// K1: bitonic argsort by descending score (stable-ish: index tiebreak).
// One block, 1024 threads, 64KB LDS (two 32KB arrays).
// ---------------------------------------------------------------------------
__global__ __launch_bounds__(SORT_THREADS)
void nms_sort_kernel(const float* __restrict__ scores, int n,
                     uint32_t* __restrict__ order) {
  __shared__ uint32_t ks[NPAD];
  __shared__ uint32_t kv[NPAD];
  const int t = threadIdx.x;
  for (int i = t; i < NPAD; i += SORT_THREADS) {
    if (i < n) {
      uint32_t b = __float_as_uint(scores[i]);
      // monotonic total-order transform, then invert for descending
      b = (b & 0x80000000u) ? ~b : (b | 0x80000000u);
      ks[i] = ~b;
      kv[i] = (uint32_t)i;
    } else {
      ks[i] = 0xFFFFFFFFu;   // pads sort last
      kv[i] = 0xFFFFFFFFu;
    }
  }
  __syncthreads();
  for (int k = 2; k <= NPAD; k <<= 1) {
    for (int j = k >> 1; j > 0; j >>= 1) {
      for (int i = t; i < NPAD; i += SORT_THREADS) {
        int ij = i ^ j;
        if (ij > i) {
          bool up = ((i & k) == 0);
          uint32_t sa = ks[i], sb = ks[ij];
          uint32_t va = kv[i], vb = kv[ij];
          bool agtb = (sa > sb) || (sa == sb && va > vb);
          if (agtb == up) { ks[i] = sb; ks[ij] = sa; kv[i] = vb; kv[ij] = va; }
        }
      }
      __syncthreads();
    }
  }
  for (int i = t; i < NPAD; i += SORT_THREADS) order[i] = kv[i];
}

// ---------------------------------------------------------------------------
// K2: gather boxes/areas into sorted order, zero-pad [n, NPAD).
// ---------------------------------------------------------------------------
__global__ void nms_gather_kernel(const float* __restrict__ boxes,
                                  const uint32_t* __restrict__ order, int n,
                                  float4* __restrict__ sbox,
                                  float* __restrict__ sarea) {
  int s = blockIdx.x * blockDim.x + threadIdx.x;
  if (s >= NPAD) return;
  if (s < n) {
    uint32_t o = order[s];
    float x1 = boxes[o * 4 + 0], y1 = boxes[o * 4 + 1];
    float x2 = boxes[o * 4 + 2], y2 = boxes[o * 4 + 3];
    sbox[s]  = make_float4(x1, y1, x2, y2);
    sarea[s] = (x2 - x1) * (y2 - y1);
  } else {
    sbox[s]  = make_float4(0.f, 0.f, 0.f, 0.f);   // inter==0 with any real box
    sarea[s] = 0.f;
  }
}

// ---------------------------------------------------------------------------
// K3: suppression bitmask. One wave32 per 16-row x 32-col tile, 8 tiles/block.
// Pairwise area-sum via V_WMMA_F32_16X16X4_F32 (rank-2: [a_i,1]x[1;a_j]).
// C-layout: VGPR r, lane l -> M = r + 8*(l/16), N = l%16.
// Row boxes fetched with broadcast global_load_b128 (no LDS permutes).
// ---------------------------------------------------------------------------
__global__ __launch_bounds__(256)
void nms_mask_kernel(const float4* __restrict__ sbox,
                     const float* __restrict__ sarea, int n,
                     uint32_t* __restrict__ mask, int ntiles) {
  const int tile = blockIdx.x * 8 + (threadIdx.x >> 5);   // wave-uniform
  if (tile >= ntiles) return;                              // uniform exit
  const int l  = threadIdx.x & 31;   // lane 0..31
  const int jw = tile & (NW - 1);    // word tile: 32 columns (NW is pow2)
  const int it = tile >> 8;          // row tile: 16 rows   (NW == 256)
  const int i0 = it * 16;
  const int j0 = jw * 32;
  const int ln = l & 15;
  const bool lo = (l < 16);
  const int h8 = lo ? 0 : 8;

  // column boxes for the two 16-wide halves
  float4 cbA = sbox[j0 + ln];
  float4 cbB = sbox[j0 + 16 + ln];
  float  caA = sarea[j0 + ln];
  float  caB = sarea[j0 + 16 + ln];
  float  rowArea = sarea[i0 + ln];

  // A (16x4): lanes 0-15 hold {K0,K1}={area_row,1}; lanes 16-31 {K2,K3}={0,0}
  // B (4x16): VGPR r, lane l -> K = r + 2*(l/16), N = l%16
  v2f a  = { lo ? rowArea : 0.f, lo ? 1.f : 0.f };
  v2f bA = { lo ? 1.f : 0.f,     lo ? caA : 0.f };
  v2f bB = { lo ? 1.f : 0.f,     lo ? caB : 0.f };
  v8f z  = {};
  // D[M][N] = area_row[M] + area_col[N]
  v8f sumA = __builtin_amdgcn_wmma_f32_16x16x4_f32(false, a, false, bA,
                                                   (short)0, z, false, false);
  v8f sumB = __builtin_amdgcn_wmma_f32_16x16x4_f32(false, a, false, bB,
                                                   (short)0, z, false, false);

  uint32_t balA[8], balB[8];
#pragma unroll
  for (int r = 0; r < 8; ++r) {
    // row box for this lane's half: only 2 distinct addresses per wave ->
    // broadcast-friendly global_load_b128, no ds_bpermute chains
    const float4 rbr = sbox[i0 + r + h8];
    const int ii = i0 + r + h8;
    {
      float w = fminf(rbr.z, cbA.z) - fmaxf(rbr.x, cbA.x);
      float h = fminf(rbr.w, cbA.w) - fmaxf(rbr.y, cbA.y);
      float inter = fmaxf(w, 0.f) * fmaxf(h, 0.f);
      int jj = j0 + ln;
      // iou > T  <=>  inter*(1+T) > T*(area_i+area_j)   (all terms >= 0)
      bool pred = (inter * (1.0f + TRESH) > TRESH * sumA[r]) &&
                  (jj > ii) && (jj < n);
      balA[r] = __builtin_amdgcn_ballot_w32(pred);
    }
    {
      float w = fminf(rbr.z, cbB.z) - fmaxf(rbr.x, cbB.x);
      float h = fminf(rbr.w, cbB.w) - fmaxf(rbr.y, cbB.y);
      float inter = fmaxf(w, 0.f) * fmaxf(h, 0.f);
      int jj = j0 + 16 + ln;
      bool pred = (inter * (1.0f + TRESH) > TRESH * sumB[r]) &&
                  (jj > ii) && (jj < n);
      balB[r] = __builtin_amdgcn_ballot_w32(pred);
    }
  }

  // assemble one u32 word per row (bit j-j0); ballots are wave-uniform
  uint32_t w = 0;
#pragma unroll
  for (int m = 0; m < 16; ++m) {
    uint32_t cand;
    if (m < 8) cand = (balA[m] & 0xFFFFu) | ((balB[m] & 0xFFFFu) << 16);
    else       cand = (balA[m - 8] >> 16) | (balB[m - 8] & 0xFFFF0000u);
    w = (ln == m) ? cand : w;
  }
  if (lo && (i0 + ln) < n) mask[(size_t)(i0 + ln) * NW + jw] = w;
}

// ---------------------------------------------------------------------------
// K4: serial greedy reduce. Single wave32; whole 8192-bit suppressed bitmap
// lives in registers (lane l owns words l*8 .. l*8+7). Prefetch next row.
// ---------------------------------------------------------------------------
__global__ __launch_bounds__(32)
void nms_reduce_kernel(const uint32_t* __restrict__ mask, int n,
                       uint32_t* __restrict__ keepw) {
  const int l = threadIdx.x;
  uint32_t S[8] = {0, 0, 0, 0, 0, 0, 0, 0};
  for (int i = 0; i < n; ++i) {
    int wi  = i >> 5;
    int bit = i & 31;
    int srcLane = wi >> 3;       // uniform
    int reg     = wi & 7;        // uniform
    uint32_t sv = 0;
#pragma unroll
    for (int r = 0; r < 8; ++r) sv = (reg == r) ? S[r] : sv;
    sv = __shfl(sv, srcLane, 32);
    if (i + 1 < n)
      __builtin_prefetch(mask + (size_t)(i + 1) * NW + l * 8, 0, 3);
    if (!((sv >> bit) & 1u)) {   // row i kept: OR its suppression row
      const uint4* row = (const uint4*)(mask + (size_t)i * NW);
      uint4 aa = row[l * 2];
      uint4 bb = row[l * 2 + 1];
      S[0] |= aa.x; S[1] |= aa.y; S[2] |= aa.z; S[3] |= aa.w;
      S[4] |= bb.x; S[5] |= bb.y; S[6] |= bb.z; S[7] |= bb.w;
    }
  }
#pragma unroll
  for (int r = 0; r < 8; ++r) keepw[l * 8 + r] = ~S[r];
}

// ---------------------------------------------------------------------------
// K5: scatter keep (sorted order) back to original order as float mask.
// ---------------------------------------------------------------------------
__global__ void nms_scatter_kernel(const uint32_t* __restrict__ keepw,
                                   const uint32_t* __restrict__ order, int n,
                                   float* __restrict__ m) {
  int s = blockIdx.x * blockDim.x + threadIdx.x;
  if (s >= n) return;
  uint32_t k = (keepw[s >> 5] >> (s & 31)) & 1u;
  m[order[s]] = (float)k;
}

// ---------------------------------------------------------------------------
// K6: write packed [N, 69] = [boxes | score | extra] * mask
// ---------------------------------------------------------------------------
__global__ void nms_pack_kernel(const float* __restrict__ boxes,
                                const float* __restrict__ scores,
                                const float* __restrict__ extra,
                                const float* __restrict__ m, int n,
                                float* __restrict__ out) {
  int t = blockIdx.x * blockDim.x + threadIdx.x;
  int total = n * 69;
  if (t >= total) return;
  int nn = t / 69;
  int c  = t - nn * 69;
  float v;
  if (c < 4)       v = boxes[nn * 4 + c];
  else if (c == 4) v = scores[nn];
  else             v = extra[nn * 64 + (c - 5)];
  out[t] = v * m[nn];
}

extern "C" void kernel_launch(void* const* d_in, const int* in_sizes, int n_in,
                              void* d_out, int out_size, void* d_ws, size_t ws_size,
                              hipStream_t stream) {
  const float* boxes  = (const float*)d_in[0];
  const float* scores = (const float*)d_in[1];
  const float* extra  = (const float*)d_in[2];
  float* out = (float*)d_out;

  int n = in_sizes[0] / 4;
  if (n <= 0 || n > NPAD) return;

  // workspace layout (bytes)
  char* ws = (char*)d_ws;
  uint32_t* order = (uint32_t*)(ws);                         //  32 KB
  float4*   sbox  = (float4*)  (ws + 32768);                 // 128 KB
  float*    sarea = (float*)   (ws + 32768 + 131072);        //  32 KB
  uint32_t* mask  = (uint32_t*)(ws + 196608);                // n*NW*4 (8 MB)
  size_t mask_bytes = (size_t)n * NW * 4;
  uint32_t* keepw = (uint32_t*)(ws + 196608 + mask_bytes);   //   1 KB
  float*    m     = (float*)   (ws + 196608 + mask_bytes + NW * 4);

  nms_sort_kernel<<<1, SORT_THREADS, 0, stream>>>(scores, n, order);
  nms_gather_kernel<<<NPAD / 256, 256, 0, stream>>>(boxes, order, n, sbox, sarea);
  int ntiles = NW * ((n + 15) / 16);
  nms_mask_kernel<<<(ntiles + 7) / 8, 256, 0, stream>>>(sbox, sarea, n, mask, ntiles);
  nms_reduce_kernel<<<1, 32, 0, stream>>>(mask, n, keepw);
  nms_scatter_kernel<<<(n + 255) / 256, 256, 0, stream>>>(keepw, order, n, m);
  nms_pack_kernel<<<(n * 69 + 255) / 256, 256, 0, stream>>>(boxes, scores, extra, m, n, out);
}